// RPSRoIPool_22144851378439
// MI455X (gfx1250) — compile-verified
//
#include <hip/hip_runtime.h>
#include <hip/hip_bf16.h>

// ---------------------------------------------------------------------------
// PSRoIPool for MI455X (gfx1250, wave32).
// Kernel 1: per-(n,c)-plane 2D integral image.
//   - Plane staged into LDS by the Tensor Data Mover (tensor_load_to_lds),
//     with the D# pad feature producing the 132-float padded row stride.
//   - Cumsums done as chained V_WMMA_F32_16X16X4_F32 triangular matmuls.
// Kernel 2: per-output-element rectangle gather from the (L2-resident)
//   integral image.
// ---------------------------------------------------------------------------

typedef __attribute__((ext_vector_type(2))) float v2f;
typedef __attribute__((ext_vector_type(8))) float v8f;
typedef __attribute__((ext_vector_type(4))) unsigned int v4u;
typedef __attribute__((ext_vector_type(8))) int v8i;
typedef __attribute__((ext_vector_type(4))) int v4i;

#define PLANES   1568      // N*C = 4*392
#define HDIM     128
#define WDIM     128
#define LSTRIDE  132       // padded LDS row stride (floats): kills bank conflicts
#define PLANE_EL (HDIM * WDIM)
#define NROIS    512
#define ODIM     8
#define CCH      392       // ODIM*7*7

static __device__ __forceinline__ v8f wmma_f32_4(v2f a, v2f b, v8f c) {
    // D(16x16) += A(16x4) * B(4x16), f32.
    return __builtin_amdgcn_wmma_f32_16x16x4_f32(false, a, false, b,
                                                 (short)0, c, false, false);
}

// One workgroup (256 threads = 8 waves) per plane.
__global__ void psroi_integral_kernel(const float* __restrict__ feat,
                                      float* __restrict__ integ) {
    extern __shared__ float lds[];   // 128 rows * LSTRIDE floats = 66 KB

    const int plane = blockIdx.x;
    const int t     = threadIdx.x;
    const int wave  = t >> 5;        // 0..7
    const int lane  = t & 31;
    const int half  = lane >> 4;     // 0/1
    const int l15   = lane & 15;

    const float* fsrc = feat + (size_t)plane * PLANE_EL;
    float*       idst = integ + (size_t)plane * PLANE_EL;

#if __has_builtin(__builtin_amdgcn_tensor_load_to_lds) && \
    __has_builtin(__builtin_amdgcn_s_wait_tensorcnt)
    // ---- stage plane into LDS via the Tensor Data Mover ----
    // D# pad feature: insert 4 DWORDs of padding after every 128 DWORDs,
    // reproducing the LSTRIDE=132 layout in hardware.
    if (t == 0) {
        const unsigned long long ga = (unsigned long long)(uintptr_t)fsrc;
        const unsigned lds_base = (unsigned)(uintptr_t)lds;  // LDS byte offset

        v4u g0;
        g0.x = 1u;                                   // count=1 (valid descriptor)
        g0.y = lds_base;                             // lds_addr (bytes)
        g0.z = (unsigned)(ga & 0xFFFFFFFFu);         // global_addr[31:0]
        g0.w = (unsigned)((ga >> 32) & 0x1FFFFFFu)   // global_addr[56:32]
             | (2u << 30);                           // type = 2 ("image")

        v8i g1;
        g1[0] = (int)((2u << 16)                     // data_size = 4 bytes
                    | (1u << 20)                     // pad_enable
                    | (6u << 22)                     // pad_interval: 128 DWORDs
                    | (3u << 25));                   // pad_amount: 4 DWORDs
        g1[1] = (int)(((unsigned)WDIM & 0xFFFFu) << 16); // tensor_dim0[15:0]
        g1[2] = (int)(((unsigned)HDIM & 0xFFFFu) << 16); // dim0[31:16]=0 | tensor_dim1[15:0]
        g1[3] = (int)(((unsigned)WDIM & 0xFFFFu) << 16); // dim1[31:16]=0 | tile_dim0=128
        g1[4] = (int)((unsigned)HDIM);                   // tile_dim1=128 | tile_dim2=0
        g1[5] = (int)((unsigned)WDIM);                   // tensor_dim0_stride[31:0]=128
        g1[6] = 0;                                       // stride0[47:32] | stride1[15:0]
        g1[7] = 0;                                       // stride1[47:16]

        v4i g2 = {0, 0, 0, 0};                       // <=2D tensor: unused
        v4i g3 = {0, 0, 0, 0};
        v8i gx = {0, 0, 0, 0, 0, 0, 0, 0};           // 6-arg form extra group:
                                                     // zero = NULL descriptor

        __builtin_amdgcn_tensor_load_to_lds(g0, g1, g2, g3, gx, 0 /*cpol*/);
        __builtin_amdgcn_s_wait_tensorcnt((short)0);
    }
    __syncthreads();
#else
    // ---- fallback: stage plane into LDS manually (coalesced float4) ----
#pragma unroll
    for (int k = 0; k < 16; ++k) {
        int f4  = t + k * 256;
        int row = f4 >> 5;
        int c4  = (f4 & 31) << 2;
        float4 v = *(const float4*)(fsrc + row * WDIM + c4);
        *(float4*)(&lds[row * LSTRIDE + c4]) = v;
    }
    __syncthreads();
#endif

    // ---- Phase A: row cumsum (along W).  wave w owns rows [16w, 16w+16) ----
    // tilecum = Tile(16x16) x U,  U upper-triangular ones (U[k][n] = k<=n),
    // chained as 4 WMMAs over K.  Carry kept in C/D layout so it folds into
    // the accumulator directly.
    {
        const int r0 = wave * 16;
        v8f carry;
#pragma unroll
        for (int v = 0; v < 8; ++v) carry[v] = 0.0f;

        for (int ct = 0; ct < 8; ++ct) {
            const int arow = r0 + l15;
            const int cb   = ct * 16 + 2 * half;
            v2f a0 = *(const v2f*)(&lds[arow * LSTRIDE + cb + 0]);
            v2f a1 = *(const v2f*)(&lds[arow * LSTRIDE + cb + 4]);
            v2f a2 = *(const v2f*)(&lds[arow * LSTRIDE + cb + 8]);
            v2f a3 = *(const v2f*)(&lds[arow * LSTRIDE + cb + 12]);

            v8f d = carry;
#pragma unroll
            for (int kk = 0; kk < 4; ++kk) {
                const int kb = kk * 4 + 2 * half;   // K-row of U this lane holds
                v2f b;
                b.x = (kb     <= l15) ? 1.0f : 0.0f;
                b.y = (kb + 1 <= l15) ? 1.0f : 0.0f;
                v2f a = (kk == 0) ? a0 : (kk == 1) ? a1 : (kk == 2) ? a2 : a3;
                d = wmma_f32_4(a, b, d);
            }

            // carry[v] <- column-15 of this tile (per row, C/D layout)
#pragma unroll
            for (int v = 0; v < 8; ++v) carry[v] = __shfl(d[v], 15, 16);

            const int srow = r0 + 8 * half;
#pragma unroll
            for (int v = 0; v < 8; ++v)
                lds[(srow + v) * LSTRIDE + ct * 16 + l15] = d[v];
        }
    }
    __syncthreads();

    // ---- Phase B: column cumsum (along H). wave w owns cols [16w, 16w+16) --
    // tilecum = Lmat x Tile,  Lmat lower-triangular ones (L[m][k] = k<=m).
    {
        const int c0 = wave * 16;
        float carry = 0.0f;

        for (int rt = 0; rt < 8; ++rt) {
            const int col = c0 + l15;
            v2f bf[4];
#pragma unroll
            for (int kk = 0; kk < 4; ++kk) {
                const int rb = rt * 16 + kk * 4 + 2 * half;
                bf[kk].x = lds[(rb    ) * LSTRIDE + col];
                bf[kk].y = lds[(rb + 1) * LSTRIDE + col];
            }

            v8f d;
#pragma unroll
            for (int v = 0; v < 8; ++v) d[v] = carry;

#pragma unroll
            for (int kk = 0; kk < 4; ++kk) {
                const int kb = kk * 4 + 2 * half;
                v2f a;
                a.x = (kb     <= l15) ? 1.0f : 0.0f;   // Lmat[m=l15][kb]
                a.y = (kb + 1 <= l15) ? 1.0f : 0.0f;
                d = wmma_f32_4(a, bf[kk], d);
            }

            // carry <- row-15 of this tile at this lane's column
            // (row 15 lives in d[7] on lanes 16..31).
            carry = __shfl(d[7], 16 + l15, 32);

            const int srow = rt * 16 + 8 * half;
#pragma unroll
            for (int v = 0; v < 8; ++v)
                lds[(srow + v) * LSTRIDE + c0 + l15] = d[v];
        }
    }
    __syncthreads();

    // ---- write integral plane out (coalesced float4); stays hot in L2 ----
#pragma unroll
    for (int k = 0; k < 16; ++k) {
        int f4  = t + k * 256;
        int row = f4 >> 5;
        int c4  = (f4 & 31) << 2;
        float4 v = *(const float4*)(&lds[row * LSTRIDE + c4]);
        *(float4*)(idst + row * WDIM + c4) = v;
    }
}

// One thread per output element: idx = ((roi*8 + ctop)*7 + ph)*7 + pw.
__global__ void psroi_gather_kernel(const float* __restrict__ rois,
                                    const float* __restrict__ integ,
                                    float* __restrict__ out) {
    const int idx = blockIdx.x * 256 + threadIdx.x;   // exact: 784*256 = 200704

    const int roi  = idx / CCH;
    const int rem  = idx - roi * CCH;
    const int ctop = rem / 49;
    const int bin  = rem - ctop * 49;
    const int ph   = bin / 7;
    const int pw   = bin - ph * 7;

    const float* r = rois + roi * 5;
    const int   b    = (int)r[0];
    const float rs_w = rintf(r[1]) * 0.0625f;
    const float rs_h = rintf(r[2]) * 0.0625f;
    const float re_w = (rintf(r[3]) + 1.0f) * 0.0625f;
    const float re_h = (rintf(r[4]) + 1.0f) * 0.0625f;
    const float roi_w = fmaxf(re_w - rs_w, 0.1f);
    const float roi_h = fmaxf(re_h - rs_h, 0.1f);
    const float bin_h = roi_h * (1.0f / 7.0f);
    const float bin_w = roi_w * (1.0f / 7.0f);

    int hs = (int)floorf((float)ph * bin_h + rs_h);
    int he = (int)ceilf (((float)ph + 1.0f) * bin_h + rs_h);
    int ws = (int)floorf((float)pw * bin_w + rs_w);
    int we = (int)ceilf (((float)pw + 1.0f) * bin_w + rs_w);
    hs = min(max(hs, 0), HDIM);
    he = min(max(he, 0), HDIM);
    ws = min(max(ws, 0), WDIM);
    we = min(max(we, 0), WDIM);

    const int c = (ctop * 7 + ph) * 7 + pw;          // gh==ph, gw==pw here
    const float* p = integ + ((size_t)(b * CCH + c)) * PLANE_EL;
    __builtin_prefetch(p + (he > 0 ? (he - 1) * WDIM : 0), 0, 3);

    // padded-integral lookup: I(0, *) == I(*, 0) == 0
    auto I = [&](int y, int x) -> float {
        return (y == 0 || x == 0) ? 0.0f : p[(y - 1) * WDIM + (x - 1)];
    };

    const float rect = I(he, we) - I(hs, we) - I(he, ws) + I(hs, ws);
    const int   area = (he - hs) * (we - ws);
    const bool  empty = (he <= hs) || (we <= ws);
    out[idx] = empty ? 0.0f : rect / fmaxf((float)area, 1.0f);
}

extern "C" void kernel_launch(void* const* d_in, const int* in_sizes, int n_in,
                              void* d_out, int out_size, void* d_ws, size_t ws_size,
                              hipStream_t stream) {
    const float* features = (const float*)d_in[0];   // 4*392*128*128 f32
    const float* rois     = (const float*)d_in[1];   // 512*5 f32
    float*       out      = (float*)d_out;           // 200704 f32
    float*       integ    = (float*)d_ws;            // needs ~98 MB scratch

    const size_t ldsBytes = (size_t)HDIM * LSTRIDE * sizeof(float); // 67584

    psroi_integral_kernel<<<PLANES, 256, ldsBytes, stream>>>(features, integ);
    psroi_gather_kernel<<<(NROIS * CCH) / 256, 256, 0, stream>>>(rois, integ, out);
    (void)in_sizes; (void)n_in; (void)out_size; (void)ws_size;
}